// SortingLayer_68942815035449
// MI455X (gfx1250) — compile-verified
//
#include <hip/hip_runtime.h>

// Soft-sort (fast-soft-sort / torchsort math), REVERSE=1, REG_STRENGTH=1.0
// x: (rows, 512) f32  ->  out: (rows, 512) f32
// Per row: sort desc -> PAV non-increasing isotonic on (w - s) -> w - sol.
//
// MI455X plan: traffic is 512 KB total (~22 ns @ 23.3 TB/s) => latency-bound.
// One workgroup (256 threads = 8 wave32) per row; row staged in/out of LDS via
// the CDNA5 async global<->LDS DMA path; bitonic sort in LDS; O(n) PAV
// replaces the reference's O(n^2) min-max formulation. No GEMM => no WMMA.

#define N   512
#define TPB 256

#if defined(__has_builtin)
#if __has_builtin(__builtin_amdgcn_global_load_async_to_lds_b64)
#define HAVE_ASYNC_LOAD 1
#endif
#if __has_builtin(__builtin_amdgcn_global_store_async_from_lds_b64)
#define HAVE_ASYNC_STORE 1
#endif
#endif
#ifndef HAVE_ASYNC_LOAD
#define HAVE_ASYNC_LOAD 0
#endif
#ifndef HAVE_ASYNC_STORE
#define HAVE_ASYNC_STORE 0
#endif

typedef int v2i __attribute__((ext_vector_type(2)));
typedef __attribute__((address_space(1))) void as1_void;
typedef __attribute__((address_space(3))) void as3_void;
typedef __attribute__((address_space(1))) v2i  as1_v2i;
typedef __attribute__((address_space(3))) v2i  as3_v2i;

__device__ __forceinline__ void wait_asynccnt0() {
#if defined(__has_builtin)
#if __has_builtin(__builtin_amdgcn_s_wait_asynccnt)
  __builtin_amdgcn_s_wait_asynccnt(0);
  return;
#endif
#endif
  asm volatile("s_wait_asynccnt 0x0" ::: "memory");
}

__global__ __launch_bounds__(TPB)
void softsort512_kernel(const float* __restrict__ x, float* __restrict__ out) {
  __shared__ float s[N];     // row buffer: input -> sorted -> final result
  __shared__ float t[N];     // w - s, later overwritten with isotonic solution y
  __shared__ float bsum[N];  // PAV block sums (stack)
  __shared__ float bcnt[N];  // PAV block counts (stack)

  const int row = blockIdx.x;
  const int tid = threadIdx.x;
  const float* xr = x + (size_t)row * N;

  // ---- stage 1: row -> LDS via CDNA5 async DMA (fallback: plain loads) ----
#if HAVE_ASYNC_LOAD
  __builtin_amdgcn_global_load_async_to_lds_b64(
      (as1_v2i*)(as1_void*)(void*)(xr + 2 * tid),
      (as3_v2i*)(as3_void*)(void*)(&s[2 * tid]),
      0, 0);
  wait_asynccnt0();
#else
  s[tid]       = xr[tid];
  s[tid + TPB] = xr[tid + TPB];
#endif
  __syncthreads();

  // ---- stage 2: bitonic sort, descending (45 stages, 2 CAS per thread) ----
  for (int k = 2; k <= N; k <<= 1) {
    for (int j = k >> 1; j > 0; j >>= 1) {
      __syncthreads();
#pragma unroll
      for (int base = 0; base < N; base += TPB) {
        const int i = base + tid;
        const int p = i ^ j;
        if (p > i) {
          const float a = s[i];
          const float b = s[p];
          const bool sw = ((i & k) == 0) ? (a < b) : (a > b);
          if (sw) { s[i] = b; s[p] = a; }
        }
      }
    }
  }
  __syncthreads();

  // ---- stage 3: t = w - s, with w[i] = (N - i) / reg_strength, reg = 1 ----
#pragma unroll
  for (int base = 0; base < N; base += TPB) {
    const int i = base + tid;
    t[i] = (float)(N - i) - s[i];
  }
  __syncthreads();

  // ---- stage 4: PAV non-increasing isotonic regression on t (thread 0) ----
  // Blocks kept as (sum, count); stack top cached in registers. Merge when the
  // new (right) block mean exceeds the top (left) mean:  mS/mC > topS/topC.
  if (tid == 0) {
    int sp = 0;
    float topS = t[0], topC = 1.0f;
    for (int i = 1; i < N; ++i) {
      float mS = t[i], mC = 1.0f;
      while (topC > 0.0f && mS * topC > topS * mC) {
        mS += topS; mC += topC;                       // merge with top block
        if (sp > 0) { --sp; topS = bsum[sp]; topC = bcnt[sp]; }
        else topC = 0.0f;                              // no left block remains
      }
      if (topC > 0.0f) { bsum[sp] = topS; bcnt[sp] = topC; ++sp; }
      topS = mS; topC = mC;
    }
    bsum[sp] = topS; bcnt[sp] = topC; ++sp;

    // expand block means into y (overwrite t)
    int pos = 0;
    for (int b = 0; b < sp; ++b) {
      const float m = bsum[b] / bcnt[b];
      const int   c = (int)bcnt[b];
      for (int r = 0; r < c; ++r) t[pos++] = m;
    }
  }
  __syncthreads();

  // ---- stage 5: result = w - y, written back into LDS row buffer ----
#pragma unroll
  for (int base = 0; base < N; base += TPB) {
    const int i = base + tid;
    s[i] = (float)(N - i) - t[i];
  }
  __syncthreads();

  // ---- stage 6: LDS -> global via CDNA5 async DMA (fallback: plain stores)
  float* outr = out + (size_t)row * N;
#if HAVE_ASYNC_STORE
  __builtin_amdgcn_global_store_async_from_lds_b64(
      (as1_v2i*)(as1_void*)(void*)(outr + 2 * tid),
      (as3_v2i*)(as3_void*)(void*)(&s[2 * tid]),
      0, 0);
  wait_asynccnt0();   // S_ENDPGM implies wait-idle; explicit for clarity
#else
  outr[tid]       = s[tid];
  outr[tid + TPB] = s[tid + TPB];
#endif
}

extern "C" void kernel_launch(void* const* d_in, const int* in_sizes, int n_in,
                              void* d_out, int out_size, void* d_ws, size_t ws_size,
                              hipStream_t stream) {
  (void)n_in; (void)d_ws; (void)ws_size; (void)out_size;
  const float* x = (const float*)d_in[0];
  float* out    = (float*)d_out;
  const int rows = in_sizes[0] / N;
  if (rows <= 0) return;
  softsort512_kernel<<<dim3(rows), dim3(TPB), 0, stream>>>(x, out);
}